// ContinuousGameOfLife_6803228197003
// MI455X (gfx1250) — compile-verified
//
#include <hip/hip_runtime.h>
#include <math.h>

// Continuous Game of Life, 16 x 2048 x 2048 fp32, toroidal wrap.
// Bandwidth-bound (roofline ~22us @ 23.3 TB/s). CDNA5 path: TDM
// tensor_load_to_lds stages each halo tile into LDS (TENSORcnt);
// boundary tiles fall back to wrap-aware float4 loads.
// Rule evaluated as a single-exponential rational form:
//   E = e^{-B(a-2.5)}, k = e^{-B}
//   out = E*(1 + E*(k + c*(1-k))) / ((E+k)*(1+kE)*(1+E))
// -> 1 v_exp_f32 + 1 v_rcp_f32 + ~9 FMAs per cell (exact algebra).

#define GOL_W   2048
#define GOL_H   2048
#define TW      128            // tile width  (outputs)
#define TH      64             // tile height (outputs)
#define EW      (TW + 8)       // staged row length: 4-float halo both sides -> 136
#define ROWS    (TH + 2)       // staged rows: 66
#define NG4     (ROWS * (EW / 4))   // float4 groups per tile = 66*34 = 2244
#define RPT     (TH / 8)       // rows per thread (8 waves * 8 rows = 64)

#define C2LOG2E 14.426950408889634f      // BETA * log2(e), BETA = 10
#define KEXP    4.5399929762484854e-05f  // e^{-BETA}

typedef __attribute__((ext_vector_type(4))) unsigned int u32x4;
typedef __attribute__((ext_vector_type(8))) int          i32x8;
typedef __attribute__((ext_vector_type(4))) int          i32x4;

__device__ __forceinline__ float fast_exp2(float v) {
#if defined(__HIP_DEVICE_COMPILE__)
    return __builtin_amdgcn_exp2f(v);
#else
    return exp2f(v);
#endif
}
__device__ __forceinline__ float fast_rcp(float v) {
#if defined(__HIP_DEVICE_COMPILE__)
    return __builtin_amdgcn_rcpf(v);
#else
    return 1.0f / v;
#endif
}

__device__ __forceinline__ float gol_rule(float c, float a) {
    const float E   = fast_exp2(fmaf(-C2LOG2E, a, 2.5f * C2LOG2E)); // e^{-B(a-2.5)}
    const float t   = fmaf(c, 1.0f - KEXP, KEXP);                   // k + c(1-k)
    const float num = E * fmaf(E, t, 1.0f);
    const float den = (E + KEXP) * fmaf(KEXP, E, 1.0f) * (1.0f + E);
    return num * fast_rcp(den);
}

__global__ __launch_bounds__(256) void gol_tdm_kernel(const float* __restrict__ x,
                                                      float* __restrict__ out) {
    __shared__ __align__(16) float sm[ROWS * EW];   // 66*136*4 = 35.9 KB

    const int tid = threadIdx.x;
    const int x0  = blockIdx.x * TW;
    const int y0  = blockIdx.y * TH;
    const size_t plane = (size_t)blockIdx.z * (size_t)GOL_H * (size_t)GOL_W;

    const bool interior = (y0 >= 1) && (y0 + TH + 1 <= GOL_H) &&
                          (x0 >= 4) && (x0 + TW + 4 <= GOL_W);

#if defined(__gfx1250__)
    const bool use_tdm = interior;
#else
    const bool use_tdm = false;
#endif

    if (use_tdm) {
#if defined(__gfx1250__)
        if (tid < 32) {  // wave 0 issues the TDM descriptor (EXEC ignored by TDM)
            const unsigned lds_base =
                (unsigned)(unsigned long long)(void*)&sm[0];
            const unsigned long long ga =
                (unsigned long long)(const void*)(x + plane +
                    (size_t)(y0 - 1) * GOL_W + (size_t)(x0 - 4));

            u32x4 g0; i32x8 g1; i32x4 g2; i32x4 g3; i32x8 g4;
            // ---- D# group 0 ----
            g0[0] = 1u;                                   // count=1 valid descriptor
            g0[1] = lds_base;                             // lds_addr (bytes)
            g0[2] = (unsigned)(ga & 0xFFFFFFFFull);       // global_addr[31:0]
            g0[3] = (unsigned)((ga >> 32) & 0x1FFFFFFull) // global_addr[56:32]
                  | (2u << 30);                           // type = 2 ("image")
            // ---- D# group 1 ----
            g1[0] = (int)(2u << 16);        // workgroup_mask=0, data_size=2 (4 bytes)
            g1[1] = (int)0xFFFF0000u;       // tensor_dim0 lo16 (dim0 = 0x7FFFFFFF)
            g1[2] = (int)0xFFFF7FFFu;       // tensor_dim1 lo16 << 16 | tensor_dim0 hi16
            g1[3] = (int)(((unsigned)EW << 16) | 0x7FFFu); // tile_dim0=136 | dim1 hi16
            g1[4] = (int)ROWS;              // tile_dim1=66, tile_dim2=0
            g1[5] = (int)GOL_W;             // tensor_dim0_stride lo32 (2048 elements)
            g1[6] = 0;                      // dim0_stride hi16 | dim1_stride lo16
            g1[7] = 0;                      // dim1_stride hi32 (unused, 2D tile)
            // ---- groups 2/3 unused (<=2D tile); extra group zeroed ----
            g2[0] = 0; g2[1] = 0; g2[2] = 0; g2[3] = 0;
            g3[0] = 0; g3[1] = 0; g3[2] = 0; g3[3] = 0;
            g4[0] = 0; g4[1] = 0; g4[2] = 0; g4[3] = 0;
            g4[4] = 0; g4[5] = 0; g4[6] = 0; g4[7] = 0;

            __builtin_amdgcn_tensor_load_to_lds(g0, g1, g2, g3, g4, 0);
            __builtin_amdgcn_s_wait_tensorcnt(0);
        }
#endif
    } else {
        // Boundary tiles: cooperative wrap-aware staging. Every 4-float group
        // is contiguous & 16B aligned since W,H,x0 are multiples of 4.
#pragma unroll
        for (int k = 0; k < (NG4 + 255) / 256; ++k) {
            const int g = tid + 256 * k;
            if (g < NG4) {
                const int r  = g / (EW / 4);
                const int c4 = g % (EW / 4);
                const int gy = (y0 - 1 + r + GOL_H) % GOL_H;
                const int gx = (x0 - 4 + c4 * 4 + GOL_W) % GOL_W;
                const float4 v = *(const float4*)(x + plane + (size_t)gy * GOL_W + gx);
                *(float4*)&sm[r * EW + c4 * 4] = v;
            }
        }
    }
    __syncthreads();

    // Wave w handles rows [8w, 8w+8); lane = column group -> coalesced stores.
    // Horizontal triple sums computed once per staged row, reused vertically.
    const int lx      = (tid & 31) * 4;
    const int rowbase = (tid >> 5) * RPT;

    float4 h[RPT + 2];     // triple sums for staged rows rowbase .. rowbase+RPT+1
    float4 mid[RPT + 2];   // center values (only rows 1..RPT used; rest DCE'd)

#pragma unroll
    for (int r = 0; r < RPT + 2; ++r) {
        const float* p = &sm[(rowbase + r) * EW + lx + 3];
        const float  l = p[0], rr = p[5];
        const float4 m = *(const float4*)(p + 1);   // 16B aligned
        mid[r] = m;
        h[r].x = l   + m.x + m.y;
        h[r].y = m.x + m.y + m.z;
        h[r].z = m.y + m.z + m.w;
        h[r].w = m.z + m.w + rr;
    }

#pragma unroll
    for (int j = 0; j < RPT; ++j) {
        const float4 c = mid[j + 1];
        float4 o;
        o.x = gol_rule(c.x, h[j].x + h[j + 1].x + h[j + 2].x - c.x);
        o.y = gol_rule(c.y, h[j].y + h[j + 1].y + h[j + 2].y - c.y);
        o.z = gol_rule(c.z, h[j].z + h[j + 1].z + h[j + 2].z - c.z);
        o.w = gol_rule(c.w, h[j].w + h[j + 1].w + h[j + 2].w - c.w);
        *(float4*)(out + plane + (size_t)(y0 + rowbase + j) * GOL_W + (x0 + lx)) = o;
    }
}

extern "C" void kernel_launch(void* const* d_in, const int* in_sizes, int n_in,
                              void* d_out, int out_size, void* d_ws, size_t ws_size,
                              hipStream_t stream) {
    (void)n_in; (void)out_size; (void)d_ws; (void)ws_size;
    const float* x  = (const float*)d_in[0];
    float* out      = (float*)d_out;
    const int batch = in_sizes[0] / (GOL_H * GOL_W);   // 16
    dim3 grid(GOL_W / TW, GOL_H / TH, batch);          // 16 x 32 x 16
    gol_tdm_kernel<<<grid, 256, 0, stream>>>(x, out);
}